// Rule_6012954214678
// MI455X (gfx1250) — compile-verified
//
#include <hip/hip_runtime.h>
#include <cmath>

#ifndef __has_builtin
#define __has_builtin(x) 0
#endif

#if __has_builtin(__builtin_amdgcn_global_load_async_to_lds_b32)
#define USE_ASYNC_LDS 1
#else
#define USE_ASYNC_LDS 0
#endif

#define TS 16
#define HTS (TS + 2)
#define TILE_ELEMS (HTS * HTS)
#define NTHREADS (TS * TS)

__device__ __forceinline__ float dirx(int d) {
  constexpr float v[9] = {0.70710678118654752f, 0.f, -0.70710678118654752f,
                          1.f, 0.f, -1.f,
                          0.70710678118654752f, 0.f, -0.70710678118654752f};
  return v[d];
}
__device__ __forceinline__ float diry(int d) {
  constexpr float v[9] = {0.70710678118654752f, 1.f, 0.70710678118654752f,
                          0.f, 0.f, 0.f,
                          -0.70710678118654752f, -1.f, -0.70710678118654752f};
  return v[d];
}

// Async global -> LDS copy (CDNA5 GLOBAL_LOAD_ASYNC_TO_LDS_B32, ASYNCcnt),
// with a plain-copy fallback if the builtin is absent on this toolchain.
__device__ __forceinline__ void gl_async_f32(const float* g, float* l) {
#if USE_ASYNC_LDS
  __builtin_amdgcn_global_load_async_to_lds_b32(
      (__attribute__((address_space(1))) int*)g,
      (__attribute__((address_space(3))) int*)l, 0, 0);
#else
  *l = *g;
#endif
}

__device__ __forceinline__ void async_wait_all() {
#if USE_ASYNC_LDS
#if __has_builtin(__builtin_amdgcn_s_wait_asynccnt)
  __builtin_amdgcn_s_wait_asynccnt(0);
#else
  asm volatile("s_wait_asynccnt 0" ::: "memory");
#endif
#endif
}

// ---------------- Pass 1: conv + force + velocity + softmax stats ------------
__global__ void __launch_bounds__(NTHREADS)
pass1_kernel(const float* __restrict__ mass, const float* __restrict__ mom,
             const float* __restrict__ force, const float* __restrict__ A,
             const float* __restrict__ ker,
             float* __restrict__ out_force,
             float* __restrict__ ws_m0, float* __restrict__ ws_m1,
             float* __restrict__ ws_vx, float* __restrict__ ws_vy,
             float* __restrict__ ws_mx, float* __restrict__ ws_rz,
             float* __restrict__ partialA, int H, int W) {
  __shared__ float sm[TILE_ELEMS];
  __shared__ float sa[TILE_ELEMS];
  __shared__ float red[NTHREADS];

  const int tx = threadIdx.x, ty = threadIdx.y;
  const int tid = ty * TS + tx;
  const int bx = blockIdx.x * TS, by = blockIdx.y * TS;
  const int N = H * W;

  for (int i = tid; i < TILE_ELEMS; i += NTHREADS) {
    const int ly = i / HTS, lx = i - ly * HTS;
    int gy = by + ly - 1;
    if (gy < 0) gy += H; else if (gy >= H) gy -= H;
    int gx = bx + lx - 1;
    if (gx < 0) gx += W; else if (gx >= W) gx -= W;
    const int g = gy * W + gx;
    gl_async_f32(mass + g, &sm[i]);
    gl_async_f32(A + g, &sa[i]);
  }
  async_wait_all();
  __syncthreads();

  // 3x3 cross-correlation of s = A*mass, 4 output channels
  float nf0 = 0.f, nf1 = 0.f, nf2 = 0.f, nf3 = 0.f;
#pragma unroll
  for (int ky = 0; ky < 3; ++ky) {
#pragma unroll
    for (int kx = 0; kx < 3; ++kx) {
      const int li = (ty + ky) * HTS + (tx + kx);
      const float sv = sm[li] * sa[li];
      const int t = ky * 3 + kx;
      nf0 = fmaf(ker[t], sv, nf0);
      nf1 = fmaf(ker[9 + t], sv, nf1);
      nf2 = fmaf(ker[18 + t], sv, nf2);
      nf3 = fmaf(ker[27 + t], sv, nf3);
    }
  }

  // stable 2-way argsort selection (matches jnp.argsort + take_along_axis)
  const float ns0 = fabsf(nf0) + fabsf(nf2);
  const float ns1 = fabsf(nf1) + fabsf(nf3);
  const bool hi = (ns1 >= ns0);
  const float zx = hi ? nf1 : nf0;
  const float zy = hi ? nf2 : nf1;

  const int g = (by + ty) * W + (bx + tx);

  const float f0 = force[g], f1 = force[N + g];
  const float fn0 = 0.9f * f0 + 0.1f * (zx - f0);
  const float fn1 = 0.9f * f1 + 0.1f * (zy - f1);
  out_force[g] = fn0;
  out_force[N + g] = fn1;

  const float mc = sm[(ty + 1) * HTS + (tx + 1)];
  const bool empty = (mc < 1e-8f);
  const float m0 = empty ? 0.f : mom[g] + fn0;      // DT = 1
  const float m1 = empty ? 0.f : mom[N + g] + fn1;
  const float vx = empty ? 0.f : m0 / mc;
  const float vy = empty ? 0.f : m1 / mc;

  float tt[9];
#pragma unroll
  for (int d = 0; d < 9; ++d) tt[d] = 0.1f * (dirx(d) * vx + diry(d) * vy);
  float mx = tt[0];
#pragma unroll
  for (int d = 1; d < 9; ++d) mx = fmaxf(mx, tt[d]);
  float Z = 0.f;
#pragma unroll
  for (int d = 0; d < 9; ++d) Z += __expf(tt[d] - mx);

  ws_m0[g] = m0; ws_m1[g] = m1;
  ws_vx[g] = vx; ws_vy[g] = vy;
  ws_mx[g] = mx; ws_rz[g] = 1.0f / Z;

  // deterministic per-block partial sum of input mass
  red[tid] = mc;
  __syncthreads();
  for (int s = NTHREADS / 2; s > 0; s >>= 1) {
    if (tid < s) red[tid] += red[tid + s];
    __syncthreads();
  }
  if (tid == 0) partialA[blockIdx.y * gridDim.x + blockIdx.x] = red[0];
}

// ---------------- Pass 2: 9-direction gather (MOVES conv) -------------------
__global__ void __launch_bounds__(NTHREADS)
pass2_kernel(const float* __restrict__ mass,
             const float* __restrict__ ws_m0, const float* __restrict__ ws_m1,
             const float* __restrict__ ws_vx, const float* __restrict__ ws_vy,
             const float* __restrict__ ws_mx, const float* __restrict__ ws_rz,
             float* __restrict__ out_mass, float* __restrict__ out_mom,
             float* __restrict__ partialB, int H, int W) {
  __shared__ float s_ms[TILE_ELEMS], s_m0[TILE_ELEMS], s_m1[TILE_ELEMS];
  __shared__ float s_vx[TILE_ELEMS], s_vy[TILE_ELEMS];
  __shared__ float s_mx[TILE_ELEMS], s_rz[TILE_ELEMS];
  __shared__ float red[NTHREADS];

  const int tx = threadIdx.x, ty = threadIdx.y;
  const int tid = ty * TS + tx;
  const int bx = blockIdx.x * TS, by = blockIdx.y * TS;
  const int N = H * W;

  for (int i = tid; i < TILE_ELEMS; i += NTHREADS) {
    const int ly = i / HTS, lx = i - ly * HTS;
    int gy = by + ly - 1;
    if (gy < 0) gy += H; else if (gy >= H) gy -= H;
    int gx = bx + lx - 1;
    if (gx < 0) gx += W; else if (gx >= W) gx -= W;
    const int g = gy * W + gx;
    gl_async_f32(mass + g, &s_ms[i]);
    gl_async_f32(ws_m0 + g, &s_m0[i]);
    gl_async_f32(ws_m1 + g, &s_m1[i]);
    gl_async_f32(ws_vx + g, &s_vx[i]);
    gl_async_f32(ws_vy + g, &s_vy[i]);
    gl_async_f32(ws_mx + g, &s_mx[i]);
    gl_async_f32(ws_rz + g, &s_rz[i]);
  }
  async_wait_all();
  __syncthreads();

  // out[h,w] = sum_i field(h+1-r, w+1-c) * P_i(src), i = 3r+c
  float am = 0.f, a0 = 0.f, a1 = 0.f;
#pragma unroll
  for (int i = 0; i < 9; ++i) {
    const int r = i / 3, c = i % 3;
    const int li = (ty + 2 - r) * HTS + (tx + 2 - c);
    const float td = 0.1f * (dirx(i) * s_vx[li] + diry(i) * s_vy[li]);
    const float P = __expf(td - s_mx[li]) * s_rz[li];
    am = fmaf(s_ms[li], P, am);
    a0 = fmaf(s_m0[li], P, a0);
    a1 = fmaf(s_m1[li], P, a1);
  }

  const int g = (by + ty) * W + (bx + tx);
  out_mass[g] = am;            // unnormalized; pass 4 rescales
  out_mom[g] = a0;
  out_mom[N + g] = a1;

  red[tid] = am;
  __syncthreads();
  for (int s = NTHREADS / 2; s > 0; s >>= 1) {
    if (tid < s) red[tid] += red[tid + s];
    __syncthreads();
  }
  if (tid == 0) partialB[blockIdx.y * gridDim.x + blockIdx.x] = red[0];
}

// ---------------- Pass 3: final deterministic reduction ---------------------
__global__ void __launch_bounds__(256)
pass3_kernel(const float* __restrict__ pA, const float* __restrict__ pB,
             int nb, float* __restrict__ scale) {
  __shared__ float ra[256], rb[256];
  const int tid = threadIdx.x;
  float sa = 0.f, sb = 0.f;
  for (int i = tid; i < nb; i += 256) { sa += pA[i]; sb += pB[i]; }
  ra[tid] = sa; rb[tid] = sb;
  __syncthreads();
  for (int s = 128; s > 0; s >>= 1) {
    if (tid < s) { ra[tid] += ra[tid + s]; rb[tid] += rb[tid + s]; }
    __syncthreads();
  }
  if (tid == 0) scale[0] = ra[0] / rb[0];
}

// ---------------- Pass 4: rescale mass --------------------------------------
__global__ void __launch_bounds__(256)
pass4_kernel(float* __restrict__ out_mass, const float* __restrict__ scale, int N) {
  const int i = blockIdx.x * 256 + threadIdx.x;
  if (i < N) out_mass[i] *= scale[0];
}

extern "C" void kernel_launch(void* const* d_in, const int* in_sizes, int n_in,
                              void* d_out, int out_size, void* d_ws, size_t ws_size,
                              hipStream_t stream) {
  (void)n_in; (void)out_size; (void)ws_size;
  const float* mass  = (const float*)d_in[0];
  const float* mom   = (const float*)d_in[1];
  const float* force = (const float*)d_in[2];
  const float* A     = (const float*)d_in[3];
  const float* ker   = (const float*)d_in[4];

  const int N = in_sizes[0];
  int H = (int)lround(sqrt((double)N));
  if (H * H != N) H = 2048;
  const int W = H;

  float* out       = (float*)d_out;
  float* out_mass  = out;                  // N
  float* out_mom   = out + N;              // 2N
  float* out_force = out + 3 * (size_t)N;  // 2N

  float* ws    = (float*)d_ws;
  float* ws_m0 = ws + 0 * (size_t)N;
  float* ws_m1 = ws + 1 * (size_t)N;
  float* ws_vx = ws + 2 * (size_t)N;
  float* ws_vy = ws + 3 * (size_t)N;
  float* ws_mx = ws + 4 * (size_t)N;
  float* ws_rz = ws + 5 * (size_t)N;

  const int GBX = W / TS, GBY = H / TS;
  const int NB = GBX * GBY;
  float* partialA = ws + 6 * (size_t)N;
  float* partialB = partialA + NB;
  float* scale    = partialB + NB;

  dim3 blk(TS, TS);
  dim3 grd(GBX, GBY);

  pass1_kernel<<<grd, blk, 0, stream>>>(mass, mom, force, A, ker, out_force,
                                        ws_m0, ws_m1, ws_vx, ws_vy, ws_mx,
                                        ws_rz, partialA, H, W);
  pass2_kernel<<<grd, blk, 0, stream>>>(mass, ws_m0, ws_m1, ws_vx, ws_vy,
                                        ws_mx, ws_rz, out_mass, out_mom,
                                        partialB, H, W);
  pass3_kernel<<<1, 256, 0, stream>>>(partialA, partialB, NB, scale);
  pass4_kernel<<<(N + 255) / 256, 256, 0, stream>>>(out_mass, scale, N);
}